// PTV3_Pooling_52965536694759
// MI455X (gfx1250) — compile-verified
//
#include <hip/hip_runtime.h>
#include <hip/hip_bf16.h>

typedef __attribute__((ext_vector_type(2))) float v2f;
typedef __attribute__((ext_vector_type(8))) float v8f;

#define LN_EPS 1e-5f
#define WT_STRIDE 68            // LDS stride in floats: bank-conflict-free, 8B-aligned pairs
#define ENC_NEG_INF (int)0x807fffff  // monotone-int encoding of -inf

// Order-preserving float<->int mapping: f32 total order == i32 total order.
__device__ __forceinline__ int enc_f32(float f) {
    int i = __float_as_int(f);
    return i >= 0 ? i : (i ^ 0x7fffffff);
}
__device__ __forceinline__ float dec_f32(int i) {
    return __int_as_float(i >= 0 ? i : (i ^ 0x7fffffff));
}

__device__ __forceinline__ v8f wmma_f32_4(v2f a, v2f b, v8f c) {
    return __builtin_amdgcn_wmma_f32_16x16x4_f32(
        false, a, false, b, (short)0, c, false, false);
}

// ---------------------------------------------------------------------------
// Kernel 0: initialize pooled buffer (d_out viewed as int) to enc(-inf)
// ---------------------------------------------------------------------------
__global__ void __launch_bounds__(256)
init_pool_kernel(int* __restrict__ out_enc, int n_ints) {
    int idx = blockIdx.x * blockDim.x + threadIdx.x;
    int i4 = idx * 4;
    if (i4 + 3 < n_ints) {
        *(int4*)(out_enc + i4) =
            make_int4(ENC_NEG_INF, ENC_NEG_INF, ENC_NEG_INF, ENC_NEG_INF);
    } else {
        for (int i = i4; i < n_ints; ++i) out_enc[i] = ENC_NEG_INF;
    }
}

// ---------------------------------------------------------------------------
// Kernel 1: fused (feats @ W + b) via V_WMMA_F32_16X16X4_F32 + scatter-max.
// One wave handles a 32-row tile (two 16-row M-subtiles sharing B fragments)
// and all 64 output channels: 128 WMMAs per wave, each LDS B-read feeds two.
// ---------------------------------------------------------------------------
__global__ void __launch_bounds__(256)
linear_scatter_max_kernel(const float* __restrict__ feats,
                          const float* __restrict__ W,
                          const float* __restrict__ bias,
                          const int*   __restrict__ pool_ids,
                          int*         __restrict__ out_enc,
                          int n_fine, int n_tiles32)
{
    __shared__ float ldsWt[64 * WT_STRIDE];   // W transposed: Wt[col][k]

    const int tid = threadIdx.x;
    // Stage W (64x64) transposed into LDS, coalesced global reads.
    for (int i = tid; i < 64 * 64; i += 256) {
        int k = i >> 6, c = i & 63;
        ldsWt[c * WT_STRIDE + k] = W[i];
    }
    __syncthreads();

    const int wave = tid >> 5;
    const int lane = tid & 31;
    const int half = lane >> 4;     // 0: lanes 0-15, 1: lanes 16-31
    const int l16  = lane & 15;
    const int tile = blockIdx.x * 8 + wave;
    if (tile >= n_tiles32) return;
    const int row0 = tile * 32;          // subtile 0 base row
    const int row1 = row0 + 16;          // subtile 1 base row
    const bool full = (row0 + 32 <= n_fine);   // wave-uniform fast-path predicate

    // A fragments for all 16 K-steps, both M-subtiles (16x4 f32 tiles):
    // lane<16 holds (row=l16, K={4k,4k+1}); lane>=16 holds (row=l16, K={4k+2,4k+3})
    v2f afrag0[16], afrag1[16];
    {
        int r0 = row0 + l16; if (r0 >= n_fine) r0 = n_fine - 1;   // clamp (guarded below)
        int r1 = row1 + l16; if (r1 >= n_fine) r1 = n_fine - 1;
        const float* p0 = feats + (size_t)r0 * 64 + 2 * half;
        const float* p1 = feats + (size_t)r1 * 64 + 2 * half;
        #pragma unroll
        for (int ks = 0; ks < 16; ++ks) {
            afrag0[ks] = *(const v2f*)(p0 + 4 * ks);
            afrag1[ks] = *(const v2f*)(p1 + 4 * ks);
        }
    }

    // Pool-id element offsets (unsigned 32-bit so the scatter uses cheap
    // zero-extended / SADDR+voffset addressing instead of 64-bit builds).
    unsigned off0[8], off1[8];
    {
        int mb0 = row0 + half * 8;
        int mb1 = row1 + half * 8;
        #pragma unroll
        for (int r = 0; r < 8; ++r) {
            int a = mb0 + r, b = mb1 + r;
            off0[r] = (a < n_fine) ? (unsigned)pool_ids[a] * 64u : 0xffffffffu;
            off1[r] = (b < n_fine) ? (unsigned)pool_ids[b] * 64u : 0xffffffffu;
        }
    }

    #pragma unroll
    for (int nt = 0; nt < 4; ++nt) {
        const int col = nt * 16 + l16;
        const float bv = bias[col];
        v8f acc0 = {bv, bv, bv, bv, bv, bv, bv, bv};   // seed with bias
        v8f acc1 = acc0;
        const float* wt = &ldsWt[col * WT_STRIDE + 2 * half];
        #pragma unroll
        for (int ks = 0; ks < 16; ++ks) {
            // B fragment (4x16 f32): lane<16 holds (K={4k,4k+1}, N=col),
            // lane>=16 holds (K={4k+2,4k+3}, N=col); contiguous in transposed LDS.
            v2f bfrag = *(const v2f*)(wt + 4 * ks);
            acc0 = wmma_f32_4(afrag0[ks], bfrag, acc0);   // independent pair:
            acc1 = wmma_f32_4(afrag1[ks], bfrag, acc1);   // no D->A/B hazard
        }
        // Scatter-max: accX[r] is element (row = rowX + half*8 + r, channel = col)
        if (full) {
            #pragma unroll
            for (int r = 0; r < 8; ++r) {
                atomicMax(out_enc + (off0[r] + (unsigned)col), enc_f32(acc0[r]));
                atomicMax(out_enc + (off1[r] + (unsigned)col), enc_f32(acc1[r]));
            }
        } else {
            #pragma unroll
            for (int r = 0; r < 8; ++r) {
                if (off0[r] != 0xffffffffu)
                    atomicMax(out_enc + (off0[r] + (unsigned)col), enc_f32(acc0[r]));
                if (off1[r] != 0xffffffffu)
                    atomicMax(out_enc + (off1[r] + (unsigned)col), enc_f32(acc1[r]));
            }
        }
    }
}

// ---------------------------------------------------------------------------
// Kernel 2: per-coarse-row LayerNorm + exact (erf) GELU, in place over d_out.
// One wave per row; 64 channels -> 2 per lane; butterfly shuffle reductions.
// ---------------------------------------------------------------------------
__global__ void __launch_bounds__(256)
ln_gelu_kernel(const float* __restrict__ gamma,
               const float* __restrict__ beta,
               float* __restrict__ out,
               int n_rows)
{
    const int wave = threadIdx.x >> 5;
    const int lane = threadIdx.x & 31;
    const int row  = blockIdx.x * 8 + wave;
    if (row >= n_rows) return;

    const int* in_enc = (const int*)out;
    const unsigned base = (unsigned)row * 64u;   // 32-bit offsets: SADDR+voffset path

    float v0 = dec_f32(in_enc[base + lane]);
    float v1 = dec_f32(in_enc[base + lane + 32]);

    float s = v0 + v1;
    #pragma unroll
    for (int off = 16; off > 0; off >>= 1) s += __shfl_xor(s, off, 32);
    const float mu = s * (1.0f / 64.0f);

    const float d0 = v0 - mu, d1 = v1 - mu;
    float q = d0 * d0 + d1 * d1;
    #pragma unroll
    for (int off = 16; off > 0; off >>= 1) q += __shfl_xor(q, off, 32);
    const float inv = rsqrtf(q * (1.0f / 64.0f) + LN_EPS);

    const float y0 = d0 * inv * gamma[lane]      + beta[lane];
    const float y1 = d1 * inv * gamma[lane + 32] + beta[lane + 32];

    const float k = 0.70710678118654752f;  // 1/sqrt(2)
    out[base + lane]      = 0.5f * y0 * (1.0f + erff(y0 * k));
    out[base + lane + 32] = 0.5f * y1 * (1.0f + erff(y1 * k));
}

// ---------------------------------------------------------------------------
extern "C" void kernel_launch(void* const* d_in, const int* in_sizes, int n_in,
                              void* d_out, int out_size, void* d_ws, size_t ws_size,
                              hipStream_t stream) {
    const float* feats    = (const float*)d_in[0];  // [N, 64] f32
    const float* W        = (const float*)d_in[1];  // [64, 64] f32
    const float* bias     = (const float*)d_in[2];  // [64] f32
    const float* gamma    = (const float*)d_in[3];  // [64] f32
    const float* beta     = (const float*)d_in[4];  // [64] f32
    const int*   pool_ids = (const int*)d_in[5];    // [N] i32
    float* out = (float*)d_out;                     // [M, 64] f32 (reused as pooled buf)

    const int n_fine = in_sizes[0] / 64;            // N
    const int n_rows = out_size / 64;               // M
    const int n_ints = n_rows * 64;

    // 1) pooled buffer := enc(-inf)
    {
        int n_vec = (n_ints + 3) / 4;
        init_pool_kernel<<<(n_vec + 255) / 256, 256, 0, stream>>>((int*)out, n_ints);
    }

    // 2) fused linear projection + scatter-max (WMMA f32)
    {
        int n_tiles32 = (n_fine + 31) / 32;
        linear_scatter_max_kernel<<<(n_tiles32 + 7) / 8, 256, 0, stream>>>(
            feats, W, bias, pool_ids, (int*)out, n_fine, n_tiles32);
    }

    // 3) LayerNorm + exact GELU, in place
    {
        ln_gelu_kernel<<<(n_rows + 7) / 8, 256, 0, stream>>>(gamma, beta, out, n_rows);
    }
}